// GraphConvShared_36129264894620
// MI455X (gfx1250) — compile-verified
//
#include <hip/hip_runtime.h>
#include <hip/hip_bf16.h>

#define GCN_NROWS 50000
#define GCN_DIN   256
#define GCN_DOUT  128
#define GCN_NEDGE 1600000

typedef float v2f __attribute__((ext_vector_type(2)));
typedef float v8f __attribute__((ext_vector_type(8)));

// ---------------------------------------------------------------------------
// Dense GEMM: H[16 rows x 128 cols per block] = X[16 x 256] * W[256 x 128]
// 8 waves per block, each wave owns one 16x16 N-tile, K-loop of 64 WMMAs.
// ---------------------------------------------------------------------------
__global__ __launch_bounds__(256) void gcn_gemm_wmma(const float* __restrict__ X,
                                                     const float* __restrict__ W,
                                                     float* __restrict__ H) {
    // Padded to 260 floats/row: keeps float4 stores 16B-aligned and spreads
    // the per-row LDS banks (bank shift of 4 per row -> conflict-free b64 reads).
    __shared__ float As[16][GCN_DIN + 4];

    const int m0  = blockIdx.x * 16;
    const int tid = threadIdx.x;

    // Cooperative stage of the 16x256 A tile (4096 floats, 16 floats/thread).
    {
        const int r  = tid >> 4;          // 0..15 row
        const int c0 = (tid & 15) * 16;   // 0,16,...,240
        const float4* src = (const float4*)(X + (size_t)(m0 + r) * GCN_DIN + c0);
        float4* dst = (float4*)(&As[r][c0]);
        dst[0] = src[0];
        dst[1] = src[1];
        dst[2] = src[2];
        dst[3] = src[3];
    }
    __syncthreads();

    const int wave  = tid >> 5;        // 0..7 -> N tile
    const int lane  = tid & 31;
    const int n0    = wave * 16;
    const int mlane = lane & 15;       // row within tile (A) / col within tile (B,D)
    const int khalf = (lane >> 4) * 2; // 0 for lanes 0-15, 2 for lanes 16-31

    v8f acc = {};

#pragma unroll 8
    for (int k0 = 0; k0 < GCN_DIN; k0 += 4) {
        // A fragment (16x4 f32): VGPR0/1 hold K=khalf / khalf+1 for row mlane.
        v2f a;
        a.x = As[mlane][k0 + khalf];
        a.y = As[mlane][k0 + khalf + 1];
        // B fragment (4x16 f32): row K striped across lanes within a VGPR.
        const float* wp = W + (size_t)(k0 + khalf) * GCN_DOUT + n0 + mlane;
        v2f b;
        b.x = wp[0];
        b.y = wp[GCN_DOUT];
        acc = __builtin_amdgcn_wmma_f32_16x16x4_f32(
            /*neg_a=*/false, a, /*neg_b=*/false, b,
            /*c_mod=*/(short)0, acc, /*reuse_a=*/false, /*reuse_b=*/false);
    }

    // D layout: VGPR r -> M = r (lanes 0-15) or M = 8+r (lanes 16-31), N = mlane.
    const int mbase = m0 + ((lane >> 4) ? 8 : 0);
    float* outp = H + (size_t)mbase * GCN_DOUT + n0 + mlane;
#pragma unroll
    for (int r = 0; r < 8; ++r) {
        outp[(size_t)r * GCN_DOUT] = acc[r];
    }
}

// ---------------------------------------------------------------------------
// SpMM scatter: one wave per edge, lane owns 4 contiguous output floats.
// msgs = h[col] * val  scatter-added into out[row] with hw f32 atomics.
// ---------------------------------------------------------------------------
__global__ __launch_bounds__(256) void gcn_spmm_scatter(const int* __restrict__ idx,
                                                        const float* __restrict__ val,
                                                        const float* __restrict__ H,
                                                        float* __restrict__ Out,
                                                        int nedges) {
    const int e = blockIdx.x * 8 + (threadIdx.x >> 5);
    if (e >= nedges) return;
    const int lane = threadIdx.x & 31;

    const int   row = idx[2 * (size_t)e];
    const int   col = idx[2 * (size_t)e + 1];
    const float v   = val[e];

    const float4 hv = *(const float4*)(H + (size_t)col * GCN_DOUT + lane * 4);
    float* op = Out + (size_t)row * GCN_DOUT + lane * 4;

    unsafeAtomicAdd(op + 0, hv.x * v);
    unsafeAtomicAdd(op + 1, hv.y * v);
    unsafeAtomicAdd(op + 2, hv.z * v);
    unsafeAtomicAdd(op + 3, hv.w * v);
}

// ---------------------------------------------------------------------------
// Elementwise helpers
// ---------------------------------------------------------------------------
__global__ __launch_bounds__(256) void gcn_zero(float* __restrict__ p, int n4) {
    // n4 = number of float4's
    int i = blockIdx.x * blockDim.x + threadIdx.x;
    int stride = gridDim.x * blockDim.x;
    float4 z = {0.f, 0.f, 0.f, 0.f};
    for (; i < n4; i += stride) ((float4*)p)[i] = z;
}

__global__ __launch_bounds__(256) void gcn_relu(float* __restrict__ p, int n4) {
    int i = blockIdx.x * blockDim.x + threadIdx.x;
    int stride = gridDim.x * blockDim.x;
    for (; i < n4; i += stride) {
        float4 x = ((float4*)p)[i];
        x.x = fmaxf(x.x, 0.f);
        x.y = fmaxf(x.y, 0.f);
        x.z = fmaxf(x.z, 0.f);
        x.w = fmaxf(x.w, 0.f);
        ((float4*)p)[i] = x;
    }
}

// ---------------------------------------------------------------------------
// Launch
// ---------------------------------------------------------------------------
extern "C" void kernel_launch(void* const* d_in, const int* in_sizes, int n_in,
                              void* d_out, int out_size, void* d_ws, size_t ws_size,
                              hipStream_t stream) {
    const float* inp_s     = (const float*)d_in[0];
    const float* inp_t     = (const float*)d_in[1];
    const int*   adj_s_idx = (const int*)d_in[2];
    const float* adj_s_val = (const float*)d_in[3];
    const int*   adj_t_idx = (const int*)d_in[4];
    const float* adj_t_val = (const float*)d_in[5];
    const float* W         = (const float*)d_in[6];

    float* out_s = (float*)d_out;
    float* out_t = out_s + (size_t)GCN_NROWS * GCN_DOUT;

    float* h_s = (float*)d_ws;
    float* h_t = h_s + (size_t)GCN_NROWS * GCN_DOUT;

    const int total   = 2 * GCN_NROWS * GCN_DOUT;  // 12.8M floats
    const int total4  = total / 4;
    const int ew_grid = (total4 + 255) / 256;

    // 1) zero-init output (atomics accumulate into it)
    gcn_zero<<<ew_grid, 256, 0, stream>>>((float*)d_out, total4);

    // 2) dense GEMMs (50000 % 16 == 0 -> exactly 3125 blocks, no tail)
    const int gemm_blocks = GCN_NROWS / 16;
    gcn_gemm_wmma<<<gemm_blocks, 256, 0, stream>>>(inp_s, W, h_s);
    gcn_gemm_wmma<<<gemm_blocks, 256, 0, stream>>>(inp_t, W, h_t);

    // 3) edge scatter (8 edges per 256-thread block)
    const int sc_blocks = (GCN_NEDGE + 7) / 8;
    gcn_spmm_scatter<<<sc_blocks, 256, 0, stream>>>(adj_s_idx, adj_s_val, h_s, out_s, GCN_NEDGE);
    gcn_spmm_scatter<<<sc_blocks, 256, 0, stream>>>(adj_t_idx, adj_t_val, h_t, out_t, GCN_NEDGE);

    // 4) ReLU over the whole concatenated output
    gcn_relu<<<ew_grid, 256, 0, stream>>>((float*)d_out, total4);
}